// TemporalConsistencyLoss_42202348650720
// MI455X (gfx1250) — compile-verified
//
#include <hip/hip_runtime.h>
#include <hip/hip_bf16.h>

typedef float v2f __attribute__((ext_vector_type(2)));
typedef float v8f __attribute__((ext_vector_type(8)));

namespace {
constexpr int kB       = 64;
constexpr int kS       = 512;
constexpr int kF       = 2048;
constexpr int kF4      = kF / 4;                       // float4 columns per row = 512
constexpr int kTC      = 4;                            // t-chunks per column
constexpr int kPPC     = (kS - 1 + kTC - 1) / kTC;     // pairs per chunk = 128
constexpr int kThreads = 256;                          // 8 waves per block (wave32)
constexpr int kBlocks  = (kB * kF4 * kTC) / kThreads;  // 512 blocks
// loss = 0.2 * sum(w*d^2) / (B * F * (S-W+1)*(W-1)) ; B*F*1020 = 133693440
constexpr float kScale = 0.2f / 133693440.0f;
}

// Wave-level fp32 sum using V_WMMA_F32_16X16X4_F32.
// A = ones(16x4), B holds one partial per lane (other B slot = 0).
// Every element of D in column n equals colsum(n) of B; summing one D
// register over all 32 lanes counts each of the 16 columns twice, i.e.
// 2 * (sum of all 64 B slots) = 2 * wave sum. Layout-mapping agnostic.
__device__ __forceinline__ float wave_reduce32(float partial) {
  v2f a;  a[0] = 1.0f;   a[1] = 1.0f;
  v2f bm; bm[0] = partial; bm[1] = 0.0f;
  v8f cm = {};
  cm = __builtin_amdgcn_wmma_f32_16x16x4_f32(
      /*neg_a=*/false, a, /*neg_b=*/false, bm,
      /*c_mod=*/(short)0, cm, /*reuse_a=*/false, /*reuse_b=*/false);
  float x = cm[0];
#pragma unroll
  for (int off = 16; off >= 1; off >>= 1)
    x += __shfl_xor(x, off, 32);
  return x * 0.5f;
}

// Stage 1: each thread walks one float4 feature column over a t-chunk,
// accumulating w(t)*||x[t+1]-x[t]||^2 with single-pass register reuse.
__global__ __launch_bounds__(kThreads) void tcl_partial_kernel(
    const float* __restrict__ seq, float* __restrict__ block_sums) {
  const int tid = blockIdx.x * kThreads + threadIdx.x;
  const int f4  = tid & (kF4 - 1);   // fastest: contiguous 512B per wave load
  const int r   = tid >> 9;
  const int b   = r & (kB - 1);
  const int c   = r >> 6;            // t-chunk id, 0..3

  const int p0 = c * kPPC;
  const int p1 = (p0 + kPPC < kS - 1) ? (p0 + kPPC) : (kS - 1);

  const float4* base =
      reinterpret_cast<const float4*>(seq) + (size_t)b * kS * kF4 + f4;

  float4 prev = base[(size_t)p0 * kF4];
  float  acc  = 0.0f;
#pragma unroll 4
  for (int p = p0; p < p1; ++p) {
    float4 cur = base[(size_t)(p + 1) * kF4];
    float dx = cur.x - prev.x;
    float dy = cur.y - prev.y;
    float dz = cur.z - prev.z;
    float dw = cur.w - prev.w;
    float s  = fmaf(dx, dx, fmaf(dy, dy, fmaf(dz, dz, dw * dw)));
    float w  = (p == 0 || p == kS - 2) ? 1.0f : 2.0f;  // mult = [1,2,...,2,1]
    acc  = fmaf(w, s, acc);
    prev = cur;
  }

  __shared__ float wsum[kThreads / 32];
  const int lane = threadIdx.x & 31;
  const int wave = threadIdx.x >> 5;
  float wtot = wave_reduce32(acc);
  if (lane == 0) wsum[wave] = wtot;
  __syncthreads();
  if (threadIdx.x == 0) {
    float s = 0.0f;
#pragma unroll
    for (int i = 0; i < kThreads / 32; ++i) s += wsum[i];
    block_sums[blockIdx.x] = s;
  }
}

// Stage 2: deterministic final reduction of the 512 block partials + scaling.
__global__ __launch_bounds__(kThreads) void tcl_final_kernel(
    const float* __restrict__ block_sums, float* __restrict__ out) {
  float partial = block_sums[threadIdx.x] + block_sums[threadIdx.x + kThreads];

  __shared__ float wsum[kThreads / 32];
  const int lane = threadIdx.x & 31;
  const int wave = threadIdx.x >> 5;
  float wtot = wave_reduce32(partial);
  if (lane == 0) wsum[wave] = wtot;
  __syncthreads();
  if (threadIdx.x == 0) {
    float s = 0.0f;
#pragma unroll
    for (int i = 0; i < kThreads / 32; ++i) s += wsum[i];
    out[0] = s * kScale;
  }
}

extern "C" void kernel_launch(void* const* d_in, const int* in_sizes, int n_in,
                              void* d_out, int out_size, void* d_ws, size_t ws_size,
                              hipStream_t stream) {
  (void)in_sizes; (void)n_in; (void)out_size; (void)ws_size;
  const float* seq = (const float*)d_in[0];
  float* block_sums = (float*)d_ws;   // needs 512 floats = 2 KB
  float* out = (float*)d_out;

  tcl_partial_kernel<<<kBlocks, kThreads, 0, stream>>>(seq, block_sums);
  tcl_final_kernel<<<1, kThreads, 0, stream>>>(block_sums, out);
}